// YOLOXHeadLibtorch_63891933495887
// MI455X (gfx1250) — compile-verified
//
#include <hip/hip_runtime.h>
#include <math.h>

// ---------------- problem constants (match reference) ----------------
#define B_        16
#define N_        33600
#define C_        80
#define MAXN      100
#define PAD       112      // MAXN padded to 7*16 for WMMA tiling
#define CAP       1024     // candidate buffer (power of two for bitonic sort)
#define SCORE_THR 0.05f
#define IOU_THR   0.65f

typedef __attribute__((ext_vector_type(2))) float v2f;
typedef __attribute__((ext_vector_type(8))) float v8f;
typedef __attribute__((ext_vector_type(4))) int   v4i;
typedef __attribute__((address_space(1))) v4i*    gv4i_p;  // global int4*
typedef __attribute__((address_space(3))) v4i*    lv4i_p;  // LDS int4*

__device__ __forceinline__ float sigmoidf_(float x) {
  return 1.0f / (1.0f + expf(-x));
}

// =====================================================================
// Kernel 1: streaming score + bbox decode.  Memory-bound: ~183 MB read,
// ~8us floor at 23.3 TB/s.  sigmoid is monotonic => max/argmax on raw
// logits, one sigmoid per anchor.  b128 loads + global_prefetch_b8.
// =====================================================================
__global__ void yolox_score_decode(const float* __restrict__ cls,
                                   const float* __restrict__ bbox,
                                   const float* __restrict__ obj,
                                   const float* __restrict__ priors,
                                   float* __restrict__ sc,
                                   int* __restrict__ lab,
                                   float4* __restrict__ box) {
  int idx = blockIdx.x * blockDim.x + threadIdx.x;
  if (idx >= B_ * N_) return;
  int n = idx % N_;

  const float4* row = (const float4*)(cls + (size_t)idx * C_);
  __builtin_prefetch(row + 20, 0, 0);  // next anchor's row -> global_prefetch_b8

  float best = -INFINITY;
  int bi = 0;
#pragma unroll
  for (int k = 0; k < 20; ++k) {       // 80 floats = 20 x float4 (b128 loads)
    float4 v = row[k];
    int base = k * 4;
    if (v.x > best) { best = v.x; bi = base;     }
    if (v.y > best) { best = v.y; bi = base + 1; }
    if (v.z > best) { best = v.z; bi = base + 2; }
    if (v.w > best) { best = v.w; bi = base + 3; }
  }

  float score = sigmoidf_(best) * sigmoidf_(obj[idx]);

  float4 p = ((const float4*)priors)[n];      // [x*s, y*s, s, s]
  float4 d = ((const float4*)bbox)[idx];
  float cx = d.x * p.z + p.x;
  float cy = d.y * p.w + p.y;
  float w  = expf(d.z) * p.z;
  float h  = expf(d.w) * p.w;
  float4 bb;
  bb.x = cx - 0.5f * w;
  bb.y = cy - 0.5f * h;
  bb.z = cx + 0.5f * w;
  bb.w = cy + 0.5f * h;

  sc[idx]  = score;
  lab[idx] = bi;
  box[idx] = bb;
}

// block-wide count of valid scores strictly above t (scores live in LDS)
__device__ int count_greater(const float* s_sc, int* s_cnt, float t,
                             int tid, int nthr) {
  if (tid == 0) *s_cnt = 0;
  __syncthreads();
  int c = 0;
  for (int n = tid; n < N_; n += nthr) {
    float s = s_sc[n];
    if (s >= SCORE_THR && s > t) ++c;
  }
  atomicAdd(s_cnt, c);
  __syncthreads();
  int r = *s_cnt;
  __syncthreads();
  return r;
}

// =====================================================================
// Kernel 2: one workgroup per batch.  Scores staged once into LDS
// (134 KB of the 320 KB WGP LDS) via GLOBAL_LOAD_ASYNC_TO_LDS_B128
// (CDNA5 async path, ASYNCcnt) when the builtin is available;
// binary-search the top-100 threshold, gather+bitonic-sort candidates,
// then NMS.  The union base matrix S[i][j] = area_i + area_j is
// computed on the matrix engine with V_WMMA_F32_16X16X4_F32 via the
// outer-sum [a,1] x [[1],[a]].
// =====================================================================
__global__ void yolox_nms(const float* __restrict__ sc,
                          const int* __restrict__ lab,
                          const float4* __restrict__ box,
                          float* __restrict__ out) {
  const int b   = blockIdx.x;
  const int tid = threadIdx.x;
  const int nthr = blockDim.x;

  extern __shared__ float smemf[];
  float* s_sc   = smemf;                       // N_ floats (134.4 KB)
  float* s_iou  = s_sc + N_;                   // PAD*PAD (union -> iou, 50 KB)
  float* cs     = s_iou + PAD * PAD;           // CAP candidate scores
  int*   ci     = (int*)(cs + CAP);            // CAP candidate indices
  float* x1_    = (float*)(ci + CAP);          // PAD each
  float* y1_    = x1_ + PAD;
  float* x2_    = y1_ + PAD;
  float* y2_    = x2_ + PAD;
  float* s_area = y2_ + PAD;
  float* s_scor = s_area + PAD;
  int*   s_lab  = (int*)(s_scor + PAD);
  int*   s_keep = s_lab + PAD;
  int*   s_cnt  = s_keep + PAD;                // 2 ints

  // ---- stage this batch's scores into LDS (coalesced, once) ----
  const float* scb = sc + (size_t)b * N_;
#if __has_builtin(__builtin_amdgcn_global_load_async_to_lds_b128) && \
    __has_builtin(__builtin_amdgcn_s_wait_asynccnt)
  {
    // direct global->LDS DMA, no VGPR round-trip; tracked by ASYNCcnt
    const float4* g4 = (const float4*)scb;
    float4*       l4 = (float4*)s_sc;
    for (int i4 = tid; i4 < N_ / 4; i4 += nthr) {
      __builtin_amdgcn_global_load_async_to_lds_b128(
          (gv4i_p)(g4 + i4), (lv4i_p)(l4 + i4), 0, 0);
    }
    __builtin_amdgcn_s_wait_asynccnt(0);
  }
#else
  for (int n = tid; n < N_; n += nthr) s_sc[n] = scb[n];
#endif
  __syncthreads();

  // ---- binary search the 100th-best valid score ----
  float lo = 0.0f, hi = 1.0f;
  for (int it = 0; it < 30; ++it) {
    float mid = 0.5f * (lo + hi);
    int c = count_greater(s_sc, s_cnt, mid, tid, nthr);
    if (c >= MAXN) lo = mid; else hi = mid;
  }

  // ---- gather candidates (score > lo, valid) ----
  if (tid == 0) s_cnt[1] = 0;
  __syncthreads();
  for (int n = tid; n < N_; n += nthr) {
    float s = s_sc[n];
    if (s >= SCORE_THR && s > lo) {
      int pos = atomicAdd(&s_cnt[1], 1);
      if (pos < CAP) { cs[pos] = s; ci[pos] = n; }
    }
  }
  __syncthreads();
  int K = s_cnt[1];
  if (K > CAP) K = CAP;
  for (int i = K + tid; i < CAP; i += nthr) { cs[i] = -INFINITY; ci[i] = 0x7fffffff; }

  // ---- bitonic sort, best-first (score desc, index asc for ties) ----
  for (int k = 2; k <= CAP; k <<= 1) {
    for (int j = k >> 1; j > 0; j >>= 1) {
      __syncthreads();
      for (int i = tid; i < CAP; i += nthr) {
        int ixj = i ^ j;
        if (ixj > i) {
          float si = cs[i], sj = cs[ixj];
          int   ii = ci[i], ij = ci[ixj];
          bool up = ((i & k) == 0);
          bool swapit = up ? ((sj > si) || (sj == si && ij < ii))
                           : ((si > sj) || (si == sj && ii < ij));
          if (swapit) { cs[i] = sj; cs[ixj] = si; ci[i] = ij; ci[ixj] = ii; }
        }
      }
    }
  }
  __syncthreads();

  // ---- load per-candidate data (top-100, pad to 112) ----
  for (int t = tid; t < PAD; t += nthr) {
    if (t < MAXN && t < K) {
      int n = ci[t];
      float4 bb = box[(size_t)b * N_ + n];
      x1_[t] = bb.x; y1_[t] = bb.y; x2_[t] = bb.z; y2_[t] = bb.w;
      s_area[t] = (bb.z - bb.x) * (bb.w - bb.y);
      s_scor[t] = cs[t];
      s_lab[t]  = lab[(size_t)b * N_ + n];
      s_keep[t] = 1;
    } else {
      x1_[t] = y1_[t] = x2_[t] = y2_[t] = 0.0f;
      s_area[t] = 0.0f; s_scor[t] = 0.0f;
      s_lab[t] = -1; s_keep[t] = 0;
    }
  }
  __syncthreads();

  // ---- WMMA: S[i][j] = area_i + area_j via outer-sum on wave 0 ----
  // A (16x4 f32): lanes 0-15 -> M=lane, v0=K0, v1=K1; lanes 16-31 -> K2,K3.
  // Row M = [area_M, 1, 0, 0];  B col N = [1, area_N, 0, 0].
  if (tid < 32) {
    int lane = tid;
    bool lo16 = lane < 16;
    int lm = lane & 15;
    for (int t5 = 0; t5 < 49; ++t5) {
      int tm = t5 / 7, tn = t5 % 7;
      v2f A;  A.x  = lo16 ? s_area[tm * 16 + lm] : 0.0f;
              A.y  = lo16 ? 1.0f : 0.0f;
      v2f Bv; Bv.x = lo16 ? 1.0f : 0.0f;
              Bv.y = lo16 ? s_area[tn * 16 + lm] : 0.0f;
      v8f Cv = {0.f, 0.f, 0.f, 0.f, 0.f, 0.f, 0.f, 0.f};
      v8f D = __builtin_amdgcn_wmma_f32_16x16x4_f32(
          false, A, false, Bv, (short)0, Cv, false, false);
      int col = tn * 16 + lm;
      int rb  = tm * 16 + (lo16 ? 0 : 8);
#pragma unroll
      for (int r = 0; r < 8; ++r) s_iou[(rb + r) * PAD + col] = D[r];
    }
  }
  __syncthreads();

  // ---- iou[i][j] = inter / (S - inter + eps), in place ----
  for (int p = tid; p < MAXN * MAXN; p += nthr) {
    int i = p / MAXN, j = p % MAXN;
    float xx1 = fmaxf(x1_[i], x1_[j]);
    float yy1 = fmaxf(y1_[i], y1_[j]);
    float xx2 = fminf(x2_[i], x2_[j]);
    float yy2 = fminf(y2_[i], y2_[j]);
    float inter = fmaxf(xx2 - xx1, 0.0f) * fmaxf(yy2 - yy1, 0.0f);
    float uni = s_iou[i * PAD + j] - inter;   // a_i + a_j - inter
    s_iou[i * PAD + j] = inter / (uni + 1e-8f);
  }
  __syncthreads();

  // ---- class-aware greedy suppression ----
  for (int i = 0; i < MAXN; ++i) {
    if (tid < MAXN && tid > i) {
      if (s_keep[i] && s_keep[tid] && s_lab[tid] == s_lab[i] &&
          s_iou[i * PAD + tid] >= IOU_THR)
        s_keep[tid] = 0;
    }
    __syncthreads();
  }

  // ---- emit dets [B,100,6] then keep [B,100] (both fp32) ----
  float* keepout = out + (size_t)B_ * MAXN * 6;
  for (int t = tid; t < MAXN; t += nthr) {
    float* o = out + ((size_t)b * MAXN + t) * 6;
    bool kp = s_keep[t] != 0;
    o[0] = kp ? x1_[t] : 0.0f;
    o[1] = kp ? y1_[t] : 0.0f;
    o[2] = kp ? x2_[t] : 0.0f;
    o[3] = kp ? y2_[t] : 0.0f;
    o[4] = kp ? s_scor[t] : 0.0f;
    o[5] = kp ? (float)s_lab[t] : -1.0f;
    keepout[b * MAXN + t] = kp ? 1.0f : 0.0f;
  }
}

// =====================================================================
extern "C" void kernel_launch(void* const* d_in, const int* in_sizes, int n_in,
                              void* d_out, int out_size, void* d_ws, size_t ws_size,
                              hipStream_t stream) {
  const float* cls    = (const float*)d_in[0];  // [B,N,80]
  const float* bbox   = (const float*)d_in[1];  // [B,N,4]
  const float* obj    = (const float*)d_in[2];  // [B,N]
  const float* priors = (const float*)d_in[3];  // [N,4]
  float* outp = (float*)d_out;

  // workspace: scores | labels | boxes  (~12.3 MB total)
  char* ws = (char*)d_ws;
  float*  sc  = (float*)ws;
  int*    lb  = (int*)(ws + (size_t)B_ * N_ * 4);
  float4* bx  = (float4*)(ws + (size_t)B_ * N_ * 8);

  const int total = B_ * N_;                 // 537600
  yolox_score_decode<<<(total + 255) / 256, 256, 0, stream>>>(
      cls, bbox, obj, priors, sc, lb, bx);

  // dynamic LDS: scores(134400) + iou(50176) + cand(8192) + misc ~= 196 KB
  const size_t smem_bytes =
      (size_t)(N_ + PAD * PAD + 2 * CAP + 8 * PAD + 8) * sizeof(float);
  yolox_nms<<<B_, 256, smem_bytes, stream>>>(sc, lb, bx, outp);

  (void)in_sizes; (void)n_in; (void)out_size; (void)ws_size;
}